// G_Unpool_90709709292193
// MI455X (gfx1250) — compile-verified
//
#include <hip/hip_runtime.h>
#include <stdint.h>
#include <stddef.h>

// Problem shape from the reference: B=16, C=64, NC=12500, N=50000, K=8
#define UB    16
#define UC    64
#define UNC   12500          // cliques
#define UN    50000          // nodes
#define UROWS (UB * UC)      // 1024 (b,c) rows
#define SCAN_T 1024

// gfx1250 async global->LDS path (guarded; falls back to float4 staging)
#if defined(__has_builtin)
#  if __has_builtin(__builtin_amdgcn_global_load_async_to_lds_b128) && \
      __has_builtin(__builtin_amdgcn_s_wait_asynccnt)
#    define UNPOOL_ASYNC_LDS 1
#  endif
#endif

// ---------------- Phase A: CSR construction over node_ids ----------------

__global__ void zero_i32(int* __restrict__ p, int n) {
  int i = blockIdx.x * blockDim.x + threadIdx.x;
  if (i < n) p[i] = 0;
}

__global__ void histo_kernel(const long long* __restrict__ node_ids,
                             int* __restrict__ counts, int nedge) {
  int e = blockIdx.x * blockDim.x + threadIdx.x;
  if (e < nedge) atomicAdd(&counts[(int)node_ids[e]], 1);
}

// Single-workgroup chunked Hillis-Steele exclusive scan (deterministic).
__global__ __launch_bounds__(SCAN_T) void scan_kernel(const int* __restrict__ counts,
                                                      int* __restrict__ row_ptr, int n) {
  __shared__ int buf[SCAN_T];
  __shared__ int carry;
  const int tid = threadIdx.x;
  if (tid == 0) { carry = 0; row_ptr[0] = 0; }
  __syncthreads();
  for (int base = 0; base < n; base += SCAN_T) {
    int i = base + tid;
    buf[tid] = (i < n) ? counts[i] : 0;
    __syncthreads();
    for (int off = 1; off < SCAN_T; off <<= 1) {
      int t = (tid >= off) ? buf[tid - off] : 0;
      __syncthreads();
      buf[tid] += t;
      __syncthreads();
    }
    if (i < n) row_ptr[i + 1] = carry + buf[tid];
    __syncthreads();
    if (tid == 0) carry += buf[SCAN_T - 1];
    __syncthreads();
  }
}

__global__ void copy_i32(const int* __restrict__ src, int* __restrict__ dst, int n) {
  int i = blockIdx.x * blockDim.x + threadIdx.x;
  if (i < n) dst[i] = src[i];
}

__global__ void fill_kernel(const long long* __restrict__ node_ids,
                            int* __restrict__ cursor, int* __restrict__ csr, int nedge) {
  int e = blockIdx.x * blockDim.x + threadIdx.x;
  if (e < nedge) {
    int node = (int)node_ids[e];
    int pos = atomicAdd(&cursor[node], 1);
    csr[pos] = e;
  }
}

// Canonicalize per-node segment order (ascending edge id) -> bit-deterministic
// float sums in phase B regardless of atomic fill order; then map edge->clique.
__global__ void sortmap_kernel(const int* __restrict__ row_ptr, int* __restrict__ csr,
                               const long long* __restrict__ clique_ids, int nodes) {
  int n = blockIdx.x * blockDim.x + threadIdx.x;
  if (n >= nodes) return;
  int s = row_ptr[n], e = row_ptr[n + 1];
  for (int i = s + 1; i < e; ++i) {        // tiny segments: avg deg 2
    int key = csr[i];
    int j = i - 1;
    while (j >= s && csr[j] > key) { csr[j + 1] = csr[j]; --j; }
    csr[j + 1] = key;
  }
  for (int i = s; i < e; ++i) csr[i] = (int)clique_ids[csr[i]];
}

// ---------------- Phase B: gather (one block per (b,c) row) ----------------

__global__ __launch_bounds__(256) void unpool_kernel(const float* __restrict__ inputs,
                                                     const int* __restrict__ row_ptr,
                                                     const int* __restrict__ csr_clique,
                                                     float* __restrict__ out) {
  __shared__ __align__(16) float srow[UNC];   // 50 KB of the 320 KB WGP LDS
  const int r = blockIdx.x;                   // r = b*C + c
  const float* __restrict__ row = inputs + (size_t)r * UNC;

#ifdef UNPOOL_ASYNC_LDS
  {
    typedef int v4i __attribute__((vector_size(16)));
    typedef __attribute__((address_space(1))) v4i* gv4_p;
    typedef __attribute__((address_space(3))) v4i* lv4_p;
    gv4_p g = (gv4_p)(uintptr_t)row;                        // AS1 global src
    lv4_p l = (lv4_p)(uint32_t)(uintptr_t)(&srow[0]);       // AS3 LDS dst (low 32 bits = LDS offset)
    for (int i = threadIdx.x; i < UNC / 4; i += blockDim.x)
      __builtin_amdgcn_global_load_async_to_lds_b128(g + i, l + i, 0, 0);
    __builtin_amdgcn_s_wait_asynccnt(0);
  }
  __syncthreads();
#else
  {
    const float4* __restrict__ row4 = (const float4*)row;
    float4* s4 = (float4*)srow;
    for (int i = threadIdx.x; i < UNC / 4; i += blockDim.x) s4[i] = row4[i];
  }
  __syncthreads();
#endif

  float* __restrict__ orow = out + (size_t)r * UN;
  for (int n = threadIdx.x; n < UN; n += blockDim.x) {
    int s = row_ptr[n];
    int e = row_ptr[n + 1];
    float acc = 0.0f;
    for (int j = s; j < e; ++j) acc += srow[csr_clique[j]];  // ds_load gathers
    orow[n] = acc;                                           // coalesced stream
  }
}

// ---------------- Fallback: atomic scatter (if workspace too small) --------

__global__ void zero_f32(float* __restrict__ p, int n) {
  int i = blockIdx.x * blockDim.x + threadIdx.x;
  if (i < n) p[i] = 0.0f;
}

__global__ __launch_bounds__(256) void scatter_kernel(const float* __restrict__ inputs,
                                                      const long long* __restrict__ node_ids,
                                                      const long long* __restrict__ clique_ids,
                                                      float* __restrict__ out, int nedge) {
  int e = blockIdx.x;
  if (e >= nedge) return;
  int node = (int)node_ids[e];
  int cli  = (int)clique_ids[e];
  for (int r = threadIdx.x; r < UROWS; r += blockDim.x)
    atomicAdd(&out[(size_t)r * UN + node], inputs[(size_t)r * UNC + cli]);
}

// ---------------------------------------------------------------------------

extern "C" void kernel_launch(void* const* d_in, const int* in_sizes, int n_in,
                              void* d_out, int out_size, void* d_ws, size_t ws_size,
                              hipStream_t stream) {
  const float*     inputs     = (const float*)d_in[0];
  const long long* node_ids   = (const long long*)d_in[1];
  const long long* clique_ids = (const long long*)d_in[2];
  float*           out        = (float*)d_out;
  const int nedge = in_sizes[1];

  const size_t need = ((size_t)(UN + 1) + (size_t)UN + (size_t)nedge) * sizeof(int);
  if (ws_size < need) {
    // Robust fallback: zero output, then per-edge atomic scatter.
    zero_f32<<<(out_size + 255) / 256, 256, 0, stream>>>(out, out_size);
    scatter_kernel<<<nedge, 256, 0, stream>>>(inputs, node_ids, clique_ids, out, nedge);
    return;
  }

  int* ws      = (int*)d_ws;
  int* row_ptr = ws;                         // UN+1
  int* cursor  = ws + (UN + 1);              // UN   (also used as counts)
  int* csr     = ws + (UN + 1) + UN;         // nedge

  zero_i32   <<<(UN + 255) / 256,    256, 0, stream>>>(cursor, UN);
  histo_kernel<<<(nedge + 255) / 256, 256, 0, stream>>>(node_ids, cursor, nedge);
  scan_kernel <<<1, SCAN_T, 0, stream>>>(cursor, row_ptr, UN);
  copy_i32    <<<(UN + 255) / 256,    256, 0, stream>>>(row_ptr, cursor, UN);
  fill_kernel <<<(nedge + 255) / 256, 256, 0, stream>>>(node_ids, cursor, csr, nedge);
  sortmap_kernel<<<(UN + 255) / 256,  256, 0, stream>>>(row_ptr, csr, clique_ids, UN);
  unpool_kernel<<<UROWS, 256, 0, stream>>>(inputs, row_ptr, csr, out);
}